// MACE_2061584302409
// MI455X (gfx1250) — compile-verified
//
#include <hip/hip_runtime.h>
#include <cmath>
#include <complex>
#include <vector>
#include <algorithm>
#include <cstdint>

// ---------------------------------------------------------------------------
// MACE-style equivariant GNN forward for MI455X (gfx1250, wave32, WMMA).
//
// Design (roofline-driven):
//  * Node features stored [n][m][f] (f fastest) so every (n,m) column is 64
//    contiguous f32 -> all channel-mixing linears are dense 64x64 GEMMs done
//    with v_wmma_f32_16x16x4_f32 (exact fp32, native CDNA5 matrix path).
//  * The dominant GEMM (radial: E x 64 @ 64 x 1472 = ~12 GFLOP/layer) is
//    FUSED into the message kernel: 16-edge Rw tile computed by WMMA into
//    LDS (94KB of 320KB/WGP), avoiding ~770MB/layer of HBM traffic.
//  * Sender-feature tile (64KB/block) staged global->LDS with CDNA5 async
//    copies (GLOBAL_LOAD_ASYNC_TO_LDS_B128, ASYNCcnt) overlapping the WMMA
//    phase; falls back to plain loads if the builtin is absent.
//  * Sparse real-Clebsch-Gordan contractions (~350 nnz over 23 paths) are
//    VALU, table precomputed on host (deterministic constants), sorted by
//    output index so accumulators are registers (no scratch spills).
//  * Scatter-add to receivers via hardware global_atomic_add_f32.
// Workspace assumption: ws_size >= ~125 MB (6 node buffers + edge buffers).
// Assumes E % 16 == 0 and N % 16 == 0 (true for the harness: 65536 / 4096).
// ---------------------------------------------------------------------------

#define CGMAX 2048
#define MSG_LDS_FLOATS (23552 + 16384 + 256 + CGMAX + CGMAX)  // rw + hs + Y + ent + coef

typedef __attribute__((ext_vector_type(2))) float v2f;
typedef __attribute__((ext_vector_type(8))) float v8f;
typedef __attribute__((__vector_size__(4*sizeof(int)))) int i4v;

#if __has_builtin(__builtin_amdgcn_global_load_async_to_lds_b128)
#define USE_ASYNC_LDS 1
#if __has_builtin(__builtin_amdgcn_s_wait_asynccnt)
#define WAIT_ASYNC() __builtin_amdgcn_s_wait_asynccnt(0)
#else
#define WAIT_ASYNC() asm volatile("s_wait_asynccnt 0" ::: "memory")
#endif
#else
#define USE_ASYNC_LDS 0
#define WAIT_ASYNC()
#endif

// ============================ device kernels ===============================

__device__ __forceinline__ float silu_f(float x) { return x / (1.0f + expf(-x)); }

// ---- edge geometry: spherical harmonics (l<=3) + radial Bessel basis ------
__global__ __launch_bounds__(256)
void k_geom(float* __restrict__ Ysh, float* __restrict__ rbf,
            const float* __restrict__ vec, int E)
{
    int e = blockIdx.x * 256 + threadIdx.x;
    if (e >= E) return;
    float x = vec[3*e+0], y = vec[3*e+1], z = vec[3*e+2];
    float rn = sqrtf(x*x + y*y + z*z);
    float inv = 1.0f / (rn + 1e-9f);
    float ux = x*inv, uy = y*inv, uz = z*inv;
    const float s3 = 1.7320508075688772f, s15 = 3.872983346207417f;
    const float s5h = 1.118033988749895f, s15h = 1.9364916731037085f;
    const float c9 = 2.091650066335189f, c10 = 10.246950765959598f;
    const float c11 = 1.6201851746019651f, c12 = 1.3228756555322954f;
    const float c14 = 5.123475382979799f;
    float* Y = Ysh + (size_t)e*16;
    Y[0]  = 1.0f;
    Y[1]  = s3*uy;  Y[2]  = s3*uz;  Y[3]  = s3*ux;
    Y[4]  = s15*ux*uy;           Y[5]  = s15*uy*uz;
    Y[6]  = s5h*(3.0f*uz*uz - 1.0f);
    Y[7]  = s15*ux*uz;           Y[8]  = s15h*(ux*ux - uy*uy);
    Y[9]  = c9*uy*(3.0f*ux*ux - uy*uy);
    Y[10] = c10*ux*uy*uz;
    Y[11] = c11*uy*(5.0f*uz*uz - 1.0f);
    Y[12] = c12*uz*(5.0f*uz*uz - 3.0f);
    Y[13] = c11*ux*(5.0f*uz*uz - 1.0f);
    Y[14] = c14*uz*(ux*ux - uy*uy);
    Y[15] = c9*ux*(ux*ux - 3.0f*uy*uy);
    float r = rn, rp = r + 1e-9f;
    float env = 0.0f;
    if (r < 1.0f) {
        float r2 = r*r, r6 = r2*r2*r2, r7 = r6*r, r8 = r7*r;
        env = 1.0f - 28.0f*r6 + 48.0f*r7 - 21.0f*r8;
    }
    float s = 1.41421356237f * env / rp;
    float* rb = rbf + (size_t)e*8;
    #pragma unroll
    for (int n = 1; n <= 8; ++n)
        rb[n-1] = s * sinf(3.14159265358979f * (float)n * r);
}

// ---- initial embedding: h[n][m][f] = (m==0) ? embed_W[sp][f] : 0 ----------
__global__ __launch_bounds__(256)
void k_embed(float* __restrict__ h, const float* __restrict__ embW,
             const int* __restrict__ species, int N)
{
    int id = blockIdx.x * 256 + threadIdx.x;
    if (id >= N*1024) return;
    int n = id >> 10, rest = id & 1023, m = rest >> 6, f = rest & 63;
    h[id] = (m == 0) ? embW[species[n]*64 + f] : 0.0f;
}

// ---- radial hidden layer: Eh = silu(rbf @ W1), K=8 (tiny; VALU) -----------
__global__ __launch_bounds__(256)
void k_radial_hidden(float* __restrict__ Eh, const float* __restrict__ rbf,
                     const float* __restrict__ W1, int E)
{
    int id = blockIdx.x * 256 + threadIdx.x;
    if (id >= E*64) return;
    int e = id >> 6, k = id & 63;
    const float* rb = rbf + (size_t)e*8;
    float a = 0.0f;
    #pragma unroll
    for (int j = 0; j < 8; ++j) a += rb[j] * W1[j*64 + k];
    Eh[id] = silu_f(a);
}

// ---- shared-weight linear over channels (per-l 64x64 GEMM) with WMMA ------
// out[n,f,m] = scale * sum_g W[l(m)][f][g]*in[n,g,m]  (+ addv[n,f,m])
// one wave per (f-band, 16-column tile); columns = (n,m) within an l-block.
__global__ __launch_bounds__(32)
void k_lin(float* __restrict__ out, const float* __restrict__ in,
           const float* __restrict__ W, const float* __restrict__ addv,
           float scale, int nnodes)
{
    const int lane = threadIdx.x;
    const int half = lane >> 4;
    const int l15  = lane & 15;
    unsigned u = blockIdx.x;
    const int fband = u & 3;
    const int tile  = u >> 2;
    const int n16 = nnodes >> 4;
    int l, tt;
    if      (tile <     n16) { l = 0; tt = tile;           }
    else if (tile < 4 * n16) { l = 1; tt = tile -     n16; }
    else if (tile < 9 * n16) { l = 2; tt = tile - 4 * n16; }
    else                     { l = 3; tt = tile - 9 * n16; }
    const int moff[4] = {0, 1, 4, 9};
    const int div = 2*l + 1;
    const int c  = tt*16 + l15;          // per-lane column within l-block
    const int n  = c / div;
    const int ml = c - n*div;
    const int m  = moff[l] + ml;
    const float* colin = in + (size_t)n*1024 + (size_t)m*64;
    const float* wrow  = W + (size_t)l*4096 + (size_t)(fband*16 + l15)*64;
    v8f acc = {0.f,0.f,0.f,0.f,0.f,0.f,0.f,0.f};
    #pragma unroll
    for (int k4 = 0; k4 < 16; ++k4) {
        const int g = 4*k4 + 2*half;
        v2f a; a.x = wrow[g];  a.y = wrow[g+1];     // A: W rows (M=f), K=g
        v2f b; b.x = colin[g]; b.y = colin[g+1];    // B: column (n,m), K=g
        acc = __builtin_amdgcn_wmma_f32_16x16x4_f32(false, a, false, b,
                                                    (short)0, acc, false, false);
    }
    const size_t obase = (size_t)n*1024 + (size_t)m*64 + fband*16 + 8*half;
    if (addv) {
        #pragma unroll
        for (int r = 0; r < 8; ++r) out[obase + r] = scale*acc[r] + addv[obase + r];
    } else {
        #pragma unroll
        for (int r = 0; r < 8; ++r) out[obase + r] = scale*acc[r];
    }
}

// ---- species-dependent linear (per-node weights; VALU) --------------------
__global__ __launch_bounds__(64)
void k_lin_sp(float* __restrict__ out, const float* __restrict__ in,
              const float* __restrict__ W, const int* __restrict__ species)
{
    const int n = blockIdx.x, f = threadIdx.x;
    const int sp = species[n];
    const float* ib = in + (size_t)n*1024;
    float* ob = out + (size_t)n*1024;
    const int lofm[16] = {0,1,1,1,2,2,2,2,2,3,3,3,3,3,3,3};
    #pragma unroll
    for (int m = 0; m < 16; ++m) {
        const float* wr = W + (((size_t)sp*4 + lofm[m])*64 + f)*64;
        const float* ir = ib + m*64;
        float a = 0.0f;
        #pragma unroll 8
        for (int g = 0; g < 64; ++g) a += wr[g]*ir[g];
        ob[m*64 + f] = a;
    }
}

// ---- fused: Rw tile (WMMA) -> CG tensor-product message -> scatter --------
__global__ __launch_bounds__(256)
void k_msg(float* __restrict__ hacc,
           const float* __restrict__ Eh,     // [E][64]
           const float* __restrict__ W2,     // [64][1472]
           const float* __restrict__ h1,     // [N][16][64]
           const float* __restrict__ Ysh,    // [E][16]
           const int* __restrict__ senders,
           const int* __restrict__ receivers,
           const unsigned* __restrict__ ent,
           const float* __restrict__ coef,
           const unsigned* __restrict__ kstart,
           int nnz)
{
    extern __shared__ float smem[];
    float* rw        = smem;                        // 16 x 1472
    float* hsl       = smem + 23552;                // 16 x 1024 gathered senders
    float* yl        = smem + 39936;                // 16 x 16
    unsigned* entl   = (unsigned*)(smem + 40192);   // CG table
    float* coefl     = smem + 40192 + CGMAX;

    const int tid  = threadIdx.x;
    const int lane = tid & 31, wv = tid >> 5;
    const int half = lane >> 4, l15 = lane & 15;
    const int e0   = blockIdx.x << 4;

    // stage sender features (async global->LDS, ASYNCcnt), Ysh, CG table
#if USE_ASYNC_LDS
    for (int i = tid; i < 16*256; i += 256) {       // 16B granules
        int el = i >> 8, off4 = i & 255;
        int s = senders[e0 + el];
        const float* g = h1 + (size_t)s*1024 + (size_t)off4*4;
        __builtin_amdgcn_global_load_async_to_lds_b128(
            (__attribute__((address_space(1))) i4v*)g,
            (__attribute__((address_space(3))) i4v*)(hsl + (size_t)i*4),
            0, 0);
    }
#else
    for (int i = tid; i < 16*1024; i += 256) {
        int el = i >> 10, off = i & 1023;
        int s = senders[e0 + el];
        hsl[i] = h1[(size_t)s*1024 + off];
    }
#endif
    { int el = tid >> 4, j = tid & 15; yl[tid] = Ysh[(size_t)(e0 + el)*16 + j]; }
    for (int i = tid; i < nnz; i += 256) { entl[i] = ent[i]; coefl[i] = coef[i]; }

    // phase 1: Rw(16 x 1472) = Eh_tile(16 x 64) @ W2 via f32 WMMA, into LDS
    v2f aK[16];
    {
        const float* er = Eh + (size_t)(e0 + l15)*64;
        #pragma unroll
        for (int k4 = 0; k4 < 16; ++k4) {
            const int g = 4*k4 + 2*half;
            aK[k4].x = er[g]; aK[k4].y = er[g+1];
        }
    }
    for (int nt = wv; nt < 92; nt += 8) {
        const int col = nt*16 + l15;
        v8f acc = {0.f,0.f,0.f,0.f,0.f,0.f,0.f,0.f};
        #pragma unroll
        for (int k4 = 0; k4 < 16; ++k4) {
            const int g = 4*k4 + 2*half;
            v2f b; b.x = W2[(size_t)g*1472 + col];
                   b.y = W2[(size_t)(g+1)*1472 + col];
            acc = __builtin_amdgcn_wmma_f32_16x16x4_f32(false, aK[k4], false, b,
                                                        (short)0, acc, false, false);
        }
        #pragma unroll
        for (int r = 0; r < 8; ++r)
            rw[(size_t)(r + 8*half)*1472 + col] = acc[r];
    }
    WAIT_ASYNC();
    __syncthreads();

    // phase 2: per (edge, channel) sparse CG contraction, scatter to receiver
    #pragma unroll
    for (int sub = 0; sub < 2; ++sub) {
        const int el = wv*2 + sub;
        const int e  = e0 + el;
        const int rc = receivers[e];
        const float* rwE = rw  + (size_t)el*1472;
        const float* hsE = hsl + (size_t)el*1024;
        const float* yE  = yl  + el*16;
        #pragma unroll
        for (int ff = 0; ff < 2; ++ff) {
            const int f = lane + 32*ff;
            float* ob = hacc + (size_t)rc*1024 + f;
            #pragma unroll
            for (int ka = 0; ka < 16; ++ka) {
                float a = 0.0f;
                const unsigned t0 = kstart[ka], t1 = kstart[ka+1];
                for (unsigned t = t0; t < t1; ++t) {
                    const unsigned pk = entl[t];
                    const int ia = pk & 255;
                    const int ja = (pk >> 8) & 255;
                    const int p  = pk >> 24;
                    a += coefl[t] * rwE[p*64 + f] * hsE[ia*64 + f] * yE[ja];
                }
                unsafeAtomicAdd(ob + ka*64, a);
            }
        }
    }
}

// ---- product basis B2 = CG(A, A; w2) --------------------------------------
__global__ __launch_bounds__(64)
void k_cgprod2(float* __restrict__ B2, const float* __restrict__ A,
               const float* __restrict__ W2p, const int* __restrict__ species,
               const unsigned* __restrict__ ent, const float* __restrict__ coef,
               const unsigned* __restrict__ kstart)
{
    const int n = blockIdx.x, f = threadIdx.x;
    const int sp = species[n];
    const float* Ab = A + (size_t)n*1024 + f;
    const float* wb = W2p + (size_t)sp*23*64 + f;
    float* Ob = B2 + (size_t)n*1024 + f;
    #pragma unroll
    for (int ka = 0; ka < 16; ++ka) {
        float a = 0.0f;
        const unsigned t0 = kstart[ka], t1 = kstart[ka+1];
        for (unsigned t = t0; t < t1; ++t) {
            const unsigned pk = ent[t];
            const int ia = pk & 255, ja = (pk >> 8) & 255, p = pk >> 24;
            a += coef[t] * wb[p*64] * Ab[ia*64] * Ab[ja*64];
        }
        Ob[ka*64] = a;
    }
}

// ---- B3 = CG(B2, A; w3); out = A*w1 + B2 + B3 -----------------------------
__global__ __launch_bounds__(64)
void k_cgprod3(float* __restrict__ out, const float* __restrict__ B2,
               const float* __restrict__ A, const float* __restrict__ W3p,
               const float* __restrict__ W1p, const int* __restrict__ species,
               const unsigned* __restrict__ ent, const float* __restrict__ coef,
               const unsigned* __restrict__ kstart)
{
    const int n = blockIdx.x, f = threadIdx.x;
    const int sp = species[n];
    const float* Ab = A  + (size_t)n*1024 + f;
    const float* Bb = B2 + (size_t)n*1024 + f;
    const float* wb = W3p + (size_t)sp*23*64 + f;
    const float w1 = W1p[sp*64 + f];
    float* Ob = out + (size_t)n*1024 + f;
    #pragma unroll
    for (int ka = 0; ka < 16; ++ka) {
        float a = 0.0f;
        const unsigned t0 = kstart[ka], t1 = kstart[ka+1];
        for (unsigned t = t0; t < t1; ++t) {
            const unsigned pk = ent[t];
            const int ia = pk & 255, ja = (pk >> 8) & 255, p = pk >> 24;
            a += coef[t] * wb[p*64] * Bb[ia*64] * Ab[ja*64];
        }
        Ob[ka*64] = Ab[ka*64]*w1 + Bb[ka*64] + a;
    }
}

// ---- readouts -------------------------------------------------------------
__global__ __launch_bounds__(256)
void k_readout(float* __restrict__ total, const float* __restrict__ h,
               const float* __restrict__ roW, int N, int init)
{
    int n = blockIdx.x * 256 + threadIdx.x;
    if (n >= N) return;
    const float* hb = h + (size_t)n*1024;  // m = 0 block
    float a = 0.0f;
    #pragma unroll 8
    for (int f = 0; f < 64; ++f) a += hb[f]*roW[f];
    total[n] = (init ? 0.0f : total[n]) + a;
}

__global__ __launch_bounds__(256)
void k_final(float* __restrict__ out, const float* __restrict__ total,
             const float* __restrict__ h, const float* __restrict__ W1,
             const float* __restrict__ W2, int N, int haveTotal)
{
    int n = blockIdx.x * 256 + threadIdx.x;
    if (n >= N) return;
    const float* hb = h + (size_t)n*1024;
    float acc = 0.0f;
    #pragma unroll
    for (int j = 0; j < 16; ++j) {
        float t = 0.0f;
        #pragma unroll 8
        for (int f = 0; f < 64; ++f) t += hb[f]*W1[f*16 + j];
        acc += silu_f(t) * W2[j];
    }
    out[n] = (haveTotal ? total[n] : 0.0f) + acc;
}

// ===================== host: real Clebsch-Gordan tables ====================

static double factd(int n) { double r = 1.0; for (int i = 2; i <= n; ++i) r *= i; return r; }

static double su2cg(int j1, int j2, int j3, int m1, int m2, int m3)
{
    if (m1 + m2 != m3) return 0.0;
    int vmin = std::max({-j1 + j2 + m3, -j1 + m1, 0});
    int vmax = std::min({j2 + j3 + m1, j3 - j1 + j2, j3 + m3});
    double C = std::sqrt((2.0*j3 + 1.0)*factd(j3+j1-j2)*factd(j3-j1+j2)*factd(j1+j2-j3)/factd(j1+j2+j3+1))
             * std::sqrt(factd(j3+m3)*factd(j3-m3)/(factd(j1-m1)*factd(j1+m1)*factd(j2-m2)*factd(j2+m2)));
    double S = 0.0;
    for (int v = vmin; v <= vmax; ++v)
        S += (((v + j2 + m2) & 1) ? -1.0 : 1.0)
           * factd(j2+j3+m1-v)*factd(j1-m1+v)
           / (factd(v)*factd(j3-j1+j2-v)*factd(j3+m3-v)*factd(v+j1-j2-m3));
    return C*S;
}

static void qmat_build(int l, std::complex<double> q[7][7])
{
    for (int a = 0; a < 7; ++a) for (int b = 0; b < 7; ++b) q[a][b] = 0.0;
    const double is2 = 1.0/std::sqrt(2.0);
    for (int m = -l; m < 0; ++m) {
        q[l+m][l-m] = std::complex<double>(is2, 0.0);
        q[l+m][l+m] = std::complex<double>(0.0, -is2);
    }
    q[l][l] = 1.0;
    for (int m = 1; m <= l; ++m) {
        double s = (m & 1) ? -1.0 : 1.0;
        q[l+m][l+m] = std::complex<double>(s*is2, 0.0);
        q[l+m][l-m] = std::complex<double>(0.0, s*is2);
    }
    std::complex<double> ph(1.0, 0.0);                  // (-1j)^l
    switch (l & 3) {
        case 1: ph = std::complex<double>(0.0, -1.0); break;
        case 2: ph = std::complex<double>(-1.0, 0.0); break;
        case 3: ph = std::complex<double>(0.0,  1.0); break;
        default: break;
    }
    for (int a = 0; a < 7; ++a) for (int b = 0; b < 7; ++b) q[a][b] *= ph;
}

static void real_cg(int l1, int l2, int l3, double C[7][7][7])
{
    std::complex<double> q1[7][7], q2[7][7], q3[7][7];
    qmat_build(l1, q1); qmat_build(l2, q2); qmat_build(l3, q3);
    const int n1 = 2*l1 + 1, n2 = 2*l2 + 1, n3 = 2*l3 + 1;
    static double Cc[7][7][7];
    for (int a = 0; a < n1; ++a)
        for (int b = 0; b < n2; ++b)
            for (int c = 0; c < n3; ++c)
                Cc[a][b][c] = su2cg(l1, l2, l3, a - l1, b - l2, c - l3);
    for (int j = 0; j < n1; ++j)
        for (int lc = 0; lc < n2; ++lc)
            for (int nc = 0; nc < n3; ++nc) {
                std::complex<double> acc(0.0, 0.0);
                for (int a = 0; a < n1; ++a)
                    for (int b = 0; b < n2; ++b)
                        for (int c = 0; c < n3; ++c)
                            if (Cc[a][b][c] != 0.0)
                                acc += q1[a][j]*q2[b][lc]*std::conj(q3[c][nc])*Cc[a][b][c];
                C[j][lc][nc] = acc.real();
            }
}

struct CgHost {
    std::vector<unsigned> ent;
    std::vector<float> coef;
    unsigned kstart[17];
    int nnz;
};

static CgHost build_cg_host()
{
    const int moff[4] = {0, 1, 4, 9};
    struct E3 { int ka; unsigned pk; float c; };
    std::vector<E3> es;
    static double C[7][7][7];
    int p = 0;
    for (int l1 = 0; l1 <= 3; ++l1)
        for (int l2 = 0; l2 <= 3; ++l2)
            for (int l3 = std::abs(l1 - l2); l3 <= std::min(3, l1 + l2); ++l3) {
                if ((l1 + l2 + l3) % 2 != 0) continue;
                real_cg(l1, l2, l3, C);
                for (int i = 0; i < 2*l1 + 1; ++i)
                    for (int j = 0; j < 2*l2 + 1; ++j)
                        for (int k = 0; k < 2*l3 + 1; ++k) {
                            double v = C[i][j][k];
                            if (std::fabs(v) > 1e-8) {
                                E3 e; e.ka = moff[l3] + k;
                                e.pk = (unsigned)(moff[l1] + i)
                                     | ((unsigned)(moff[l2] + j) << 8)
                                     | ((unsigned)e.ka << 16)
                                     | ((unsigned)p << 24);
                                e.c = (float)v;
                                es.push_back(e);
                            }
                        }
                ++p;
            }
    std::stable_sort(es.begin(), es.end(),
                     [](const E3& a, const E3& b) { return a.ka < b.ka; });
    CgHost t;
    t.nnz = (int)std::min<size_t>(es.size(), CGMAX);
    t.ent.resize(t.nnz); t.coef.resize(t.nnz);
    for (int i = 0; i < t.nnz; ++i) { t.ent[i] = es[i].pk; t.coef[i] = es[i].c; }
    int idx = 0;
    for (int ka = 0; ka <= 16; ++ka) {
        while (idx < t.nnz && es[idx].ka < ka) ++idx;
        t.kstart[ka] = (unsigned)idx;
    }
    t.kstart[16] = (unsigned)t.nnz;
    return t;
}

static const CgHost& get_cg() { static CgHost t = build_cg_host(); return t; }

// ============================== launcher ===================================

extern "C" void kernel_launch(void* const* d_in, const int* in_sizes, int n_in,
                              void* d_out, int out_size, void* d_ws, size_t ws_size,
                              hipStream_t stream)
{
    (void)n_in; (void)out_size; (void)ws_size;
    const float* vectors  = (const float*)d_in[0];
    const int*   senders  = (const int*)  d_in[1];
    const int*   receivers= (const int*)  d_in[2];
    const int*   species  = (const int*)  d_in[3];
    const float* embW     = (const float*)d_in[4];
    const float* skipW    = (const float*)d_in[5];
    const float* lin1W    = (const float*)d_in[6];
    const float* rW1      = (const float*)d_in[7];
    const float* rW2      = (const float*)d_in[8];
    const float* lin2W    = (const float*)d_in[9];
    const float* skipFW   = (const float*)d_in[10];
    const float* pW1      = (const float*)d_in[11];
    const float* pW2      = (const float*)d_in[12];
    const float* pW3      = (const float*)d_in[13];
    const float* plinW    = (const float*)d_in[14];
    const float* roW      = (const float*)d_in[15];
    const float* mlpW1    = (const float*)d_in[16];
    const float* mlpW2    = (const float*)d_in[17];

    const int E = in_sizes[0] / 3;
    const int N = in_sizes[3];
    const int S = in_sizes[4] / 64;
    const int L = in_sizes[6] / (4*64*64);

    const CgHost& cg = get_cg();

    float* wsf = (float*)d_ws;
    size_t o = 0;
    float* Ysh = wsf + o; o += (size_t)E*16;
    float* rbf = wsf + o; o += (size_t)E*8;
    float* Eh  = wsf + o; o += (size_t)E*64;
    float* nb[6];
    for (int b = 0; b < 6; ++b) { nb[b] = wsf + o; o += (size_t)N*1024; }
    float* total = wsf + o; o += (size_t)N;
    unsigned* d_kstart = (unsigned*)(wsf + o); o += 32;
    unsigned* d_ent    = (unsigned*)(wsf + o); o += CGMAX;
    float*    d_coef   = wsf + o;              o += CGMAX;

    (void)hipMemcpyAsync(d_kstart, cg.kstart, 17*sizeof(unsigned), hipMemcpyHostToDevice, stream);
    (void)hipMemcpyAsync(d_ent, cg.ent.data(), (size_t)cg.nnz*sizeof(unsigned), hipMemcpyHostToDevice, stream);
    (void)hipMemcpyAsync(d_coef, cg.coef.data(), (size_t)cg.nnz*sizeof(float), hipMemcpyHostToDevice, stream);

    const float eps = 1.0f / sqrtf(17.0f);   // 1/sqrt(1 + AVG_NEIGH)

    k_geom<<<(E + 255)/256, 256, 0, stream>>>(Ysh, rbf, vectors, E);
    float* h = nb[0];
    k_embed<<<(N*1024 + 255)/256, 256, 0, stream>>>(h, embW, species, N);

    for (int i = 0; i < L; ++i) {
        const bool first = (i == 0), last = (i == L - 1);
        float* SC = nb[5];
        if (!first)
            k_lin_sp<<<N, 64, 0, stream>>>(SC, h, skipW + (size_t)i*S*16384, species);
        float* H1 = nb[1];
        k_lin<<<4*N, 32, 0, stream>>>(H1, h, lin1W + (size_t)i*16384, nullptr, 1.0f, N);
        k_radial_hidden<<<(E*64 + 255)/256, 256, 0, stream>>>(Eh, rbf, rW1 + (size_t)i*512, E);
        float* MACC = nb[2];
        (void)hipMemsetAsync(MACC, 0, (size_t)N*1024*sizeof(float), stream);
        k_msg<<<E/16, 256, (size_t)MSG_LDS_FLOATS*sizeof(float), stream>>>(
            MACC, Eh, rW2 + (size_t)i*64*1472, H1, Ysh, senders, receivers,
            d_ent, d_coef, d_kstart, cg.nnz);
        float* HL = nb[3];
        k_lin<<<4*N, 32, 0, stream>>>(HL, MACC, lin2W + (size_t)i*16384, nullptr, eps, N);
        float* A = HL;
        if (first) {
            k_lin_sp<<<N, 64, 0, stream>>>(nb[1], HL, skipFW, species);
            A = nb[1];
        }
        float* B2 = nb[2];  // MACC dead after lin2
        k_cgprod2<<<N, 64, 0, stream>>>(B2, A, pW2 + (size_t)i*S*1472, species,
                                        d_ent, d_coef, d_kstart);
        float* CMB = nb[4];
        k_cgprod3<<<N, 64, 0, stream>>>(CMB, B2, A, pW3 + (size_t)i*S*1472,
                                        pW1 + (size_t)i*S*64, species,
                                        d_ent, d_coef, d_kstart);
        k_lin<<<4*N, 32, 0, stream>>>(h, CMB, plinW + (size_t)i*16384,
                                      first ? nullptr : SC, 1.0f, N);
        if (!last)
            k_readout<<<(N + 255)/256, 256, 0, stream>>>(total, h, roW, N, first ? 1 : 0);
        else
            k_final<<<(N + 255)/256, 256, 0, stream>>>((float*)d_out, total, h,
                                                       mlpW1, mlpW2, N, (i > 0) ? 1 : 0);
    }
}